// CumMulSumModule_68908455297459
// MI455X (gfx1250) — compile-verified
//
#include <hip/hip_runtime.h>
#include <stdint.h>

// Problem shape (fixed by the reference: x[8,4096,1024] f32, d[1024] f32)
#define BSZ   8
#define TLEN  4096
#define CH    1024

// Decomposition
#define LCH   256                 // timesteps per chunk (per block)
#define NCH   (TLEN / LCH)        // 16 chunks -> chain depth 16
#define GCH   256                 // channels per block (== blockDim.x)
#define NGR   (CH / GCH)          // 4 channel groups
#define TS    16                  // timesteps per LDS stage
#define NSTG  (LCH / TS)          // 16 stages
#define NLD   4                   // b128 async loads per thread per stage

// Workspace layout
#define FLAGS_BYTES ((size_t)BSZ * NCH * NGR * sizeof(int))    // 2 KB
#define CARRY_OFF   4096
#define CARRY_BYTES ((size_t)BSZ * NCH * CH * sizeof(float))   // 512 KB
#define WS_NEED     (CARRY_OFF + CARRY_BYTES)

// ---- async copy of one 16x256-f32 tile (16 KB) from global to LDS ----------
// Each thread issues NLD global_load_async_to_lds_b128 (16 B per lane each),
// tracked by ASYNCcnt, retired in order. th:TH_LOAD_LU marks the line as
// last-use: after staging to LDS this x data is dead, so drop it from cache
// and keep L2 free for the phase-1 read stream of other blocks.
__device__ __forceinline__ void prefetch_stage(const float* __restrict__ xg,
                                               int c0, int stage,
                                               unsigned lbase, int tid) {
#pragma unroll
  for (int i = 0; i < NLD; ++i) {
    int q   = tid + i * 256;              // 0..1023 : 16B chunk index in tile
    int row = q >> 6;                     // 0..15   : timestep within stage
    int col = (q & 63) << 2;              // float column within 256
    const float* g = xg + (size_t)(stage * TS + row) * CH + c0 + col;
    unsigned loff = lbase + (unsigned)q * 16u;
    asm volatile("global_load_async_to_lds_b128 %0, %1, off th:TH_LOAD_LU"
                 :: "v"(loff), "v"(g) : "memory");
  }
}

__global__ __launch_bounds__(256)
void scan_chain_kernel(const float* __restrict__ x,
                       const float* __restrict__ d,
                       float* __restrict__ out,
                       int* __restrict__ flags,
                       float* __restrict__ carry) {
  __shared__ float tile[2][TS][GCH];      // 32 KB, double buffered

  const int tid = threadIdx.x;            // 0..255 == channel within group
  const int b   = blockIdx.x >> 2;        // batch
  const int g   = blockIdx.x & (NGR - 1); // channel group
  const int j   = blockIdx.y;             // chunk (predecessors launch first)
  const int c0  = g * GCH;
  const int c   = c0 + tid;
  const int t0  = j * LCH;

  const float dch = d[c];
  const float* xg = x + ((size_t)b * TLEN + t0) * CH;  // chunk row base

  // LDS byte offsets of the two buffers (low 32 bits of generic LDS ptr).
  const unsigned lbase0 = (unsigned)(uintptr_t)&tile[0][0][0];
  const unsigned lbase1 = (unsigned)(uintptr_t)&tile[1][0][0];

  // Kick off async staging of the first two phase-2 tiles immediately; they
  // fetch the same lines phase 1 is about to read.
  prefetch_stage(xg, c0, 0, lbase0, tid);
  prefetch_stage(xg, c0, 1, lbase1, tid);

  // ---- Phase 1: chunk-local affine aggregate (a = d^L, B = local scan end)
  // Deep unroll -> 16 outstanding 128B/wave loads to cover HBM latency.
  float s = 0.0f, a = 1.0f;
  {
    const float* px = xg + c;
#pragma unroll 16
    for (int t = 0; t < LCH; ++t) {
      s = __builtin_fmaf(dch, s, px[(size_t)t * CH]);
      a *= dch;                           // exact repeated product
    }
  }

  // ---- Chain link: wait for predecessor carry, publish inclusive carry
  float cprev = 0.0f;
  if (j > 0) {
    int* flg = flags + ((b * NCH + (j - 1)) * NGR + g);
    if (tid == 0) {
      while (__hip_atomic_load(flg, __ATOMIC_ACQUIRE,
                               __HIP_MEMORY_SCOPE_AGENT) == 0) {
        __builtin_amdgcn_s_sleep(2);
      }
    }
    __syncthreads();
    cprev = __hip_atomic_load(carry + ((size_t)(b * NCH + (j - 1)) * CH + c),
                              __ATOMIC_RELAXED, __HIP_MEMORY_SCOPE_AGENT);
  }
  const float cj = __builtin_fmaf(a, cprev, s);
  if (j < NCH - 1) {
    __hip_atomic_store(carry + ((size_t)(b * NCH + j) * CH + c), cj,
                       __ATOMIC_RELAXED, __HIP_MEMORY_SCOPE_AGENT);
    __threadfence();                      // carries agent-visible before flag
    __syncthreads();
    if (tid == 0) {
      __hip_atomic_store(flags + ((b * NCH + j) * NGR + g), 1,
                         __ATOMIC_RELEASE, __HIP_MEMORY_SCOPE_AGENT);
    }
  }

  // ---- Phase 2: exact sequential scan with carry, x staged via async LDS.
  // Output stores are non-temporal: y is never re-read, keep it out of L2
  // so the x stream (which IS re-read by the async stager) stays resident.
  float y = cprev;
  float* og = out + ((size_t)b * TLEN + t0) * CH + c;

  for (int k = 0; k < NSTG; ++k) {
    // Stage k's 4 per-wave async loads are done once ASYNCcnt <= 4 (the
    // in-flight stage k+1); last stage drains to 0.
    if (k < NSTG - 1) asm volatile("s_wait_asynccnt 4" ::: "memory");
    else              asm volatile("s_wait_asynccnt 0" ::: "memory");
    __syncthreads();                      // all waves' tile-k writes visible

    const float* tk = &tile[k & 1][0][tid];
#pragma unroll
    for (int r = 0; r < TS; ++r) {
      y = __builtin_fmaf(dch, y, tk[(size_t)r * GCH]);
      __builtin_nontemporal_store(y, &og[(size_t)(k * TS + r) * CH]);
    }
    __syncthreads();                      // tile-k reads done before refill
    if (k + 2 < NSTG)
      prefetch_stage(xg, c0, k + 2, (k & 1) ? lbase1 : lbase0, tid);
  }
}

// Workspace-free fallback (used only if ws_size is unexpectedly tiny):
// one thread per (batch, channel), fully sequential but coalesced.
__global__ __launch_bounds__(256)
void naive_scan_kernel(const float* __restrict__ x,
                       const float* __restrict__ d,
                       float* __restrict__ out) {
  int idx = blockIdx.x * blockDim.x + threadIdx.x;   // 0..8191
  int b = idx >> 10, c = idx & (CH - 1);
  const float* px = x + (size_t)b * TLEN * CH + c;
  float* po = out + (size_t)b * TLEN * CH + c;
  float dch = d[c], y = 0.0f;
  for (int t = 0; t < TLEN; ++t) {
    y = __builtin_fmaf(dch, y, px[(size_t)t * CH]);
    po[(size_t)t * CH] = y;
  }
}

extern "C" void kernel_launch(void* const* d_in, const int* in_sizes, int n_in,
                              void* d_out, int out_size, void* d_ws, size_t ws_size,
                              hipStream_t stream) {
  (void)in_sizes; (void)n_in; (void)out_size;
  const float* x = (const float*)d_in[0];
  const float* d = (const float*)d_in[1];
  float* out = (float*)d_out;

  if (ws_size < WS_NEED) {
    naive_scan_kernel<<<(BSZ * CH) / 256, 256, 0, stream>>>(x, d, out);
    return;
  }

  int*   flags = (int*)d_ws;
  float* carry = (float*)((char*)d_ws + CARRY_OFF);

  // Reset chain flags every call (graph-capture safe, stream-ordered).
  hipMemsetAsync(d_ws, 0, FLAGS_BYTES, stream);

  dim3 grid(BSZ * NGR, NCH, 1);           // x: independent chains, y: chunk order
  scan_chain_kernel<<<grid, GCH, 0, stream>>>(x, d, out, flags, carry);
}